// HybridQLSTM_65481071399917
// MI455X (gfx1250) — compile-verified
//
#include <hip/hip_runtime.h>
#include <hip/hip_bf16.h>
#include <math.h>

typedef float v2f __attribute__((ext_vector_type(2)));
typedef float v8f __attribute__((ext_vector_type(8)));

#define S_LEN 256
#define BATCH 128
#define EMB   256
#define NW    4
#define NGATE 16     // 4 gates * 4 wires, stacked rows [Wf;Wi;Wu;Wo]
#define FIN   260    // E + H

// ---------------------------------------------------------------------------
// Kernel 1: pre[s][b][g] = sum_e emb[sentence[s,b], e] * Wall[g][e]
// One wave per (s, 16-batch tile). V_WMMA_F32_16X16X4_F32, K = 256 (64 steps).
//   A (16x4 f32):  lane<16  -> (A[M][k], A[M][k+1]),  lane>=16 -> (A[M][k+2], A[M][k+3])
//   B (4x16 f32):  lane<16  -> (B[k][N], B[k+1][N]),  lane>=16 -> (B[k+2][N], B[k+3][N])
//   D (16x16 f32): vgpr v: lanes0-15 M=v, lanes16-31 M=v+8; N = lane&15
// ---------------------------------------------------------------------------
__global__ __launch_bounds__(32)
void qlstm_gemm_embed(const int* __restrict__ sentence,
                      const float* __restrict__ emb,
                      const float* __restrict__ Wf,
                      const float* __restrict__ Wi,
                      const float* __restrict__ Wu,
                      const float* __restrict__ Wo,
                      float* __restrict__ pre) {
  const int s    = blockIdx.y;
  const int bt   = blockIdx.x;        // 16-row batch tile
  const int lane = threadIdx.x;       // 0..31
  const int mn   = lane & 15;         // M for A-rows, N for B-cols
  const int hi   = lane >> 4;         // K sub-pair select

  // A: gathered embedding row for this lane's M
  const int tok = sentence[s * BATCH + bt * 16 + mn];
  const float* arow = emb + (long)tok * EMB + 2 * hi;

  // B: stacked gate-weight row mn of [Wf;Wi;Wu;Wo] (row stride FIN)
  const float* wbase = (mn < 8) ? ((mn < 4) ? Wf : Wi)
                                : ((mn < 12) ? Wu : Wo);
  const float* brow = wbase + (mn & 3) * FIN + 2 * hi;

  v8f acc = {};
  #pragma unroll 8
  for (int k = 0; k < EMB; k += 4) {
    v2f a = *reinterpret_cast<const v2f*>(arow + k);   // global_load_b64
    v2f b = *reinterpret_cast<const v2f*>(brow + k);   // global_load_b64 (L2-hot)
    acc = __builtin_amdgcn_wmma_f32_16x16x4_f32(
        /*neg_a=*/false, a, /*neg_b=*/false, b,
        /*c_mod=*/(short)0, acc, /*reuse_a=*/false, /*reuse_b=*/false);
  }

  float* out = pre + ((long)s * BATCH + bt * 16) * NGATE + mn;
  #pragma unroll
  for (int v = 0; v < 8; ++v) {
    const int M = v + 8 * hi;
    out[M * NGATE] = acc[v];
  }
}

// ---------------------------------------------------------------------------
// Kernel 2: sequential LSTM scan. 1 block of 128 threads, 1 thread per batch.
// Quantum layer collapsed analytically:
//   c_w = cos(z_w + theta_w);  q = {c1c2c3, c0c1, c0c1c2, c0c1c2c3}
// Final log_softmax of a constant-broadcast row == lg - (lg + ln 16).
// ---------------------------------------------------------------------------
__device__ __forceinline__ float sig_fast(float x)  { return 1.0f / (1.0f + __expf(-x)); }
__device__ __forceinline__ float tanh_fast(float x) { return 1.0f - 2.0f / (__expf(2.0f * x) + 1.0f); }

__global__ __launch_bounds__(BATCH)
void qlstm_recur(const float* __restrict__ pre,
                 const float* __restrict__ Wf, const float* __restrict__ bf,
                 const float* __restrict__ Wi, const float* __restrict__ bi,
                 const float* __restrict__ Wu, const float* __restrict__ bu,
                 const float* __restrict__ Wo, const float* __restrict__ bo,
                 const float* __restrict__ thf, const float* __restrict__ thi,
                 const float* __restrict__ thu, const float* __restrict__ tho,
                 float* __restrict__ out) {
  __shared__ float sWh[NGATE][NW];   // h-recurrence columns W[g][256..259]
  __shared__ float sB[NGATE];
  __shared__ float sTh[NGATE];

  const int b = threadIdx.x;
  if (b < NGATE) {
    const int G = b >> 2, r = b & 3;
    const float* W  = (G == 0) ? Wf  : (G == 1) ? Wi  : (G == 2) ? Wu  : Wo;
    const float* bb = (G == 0) ? bf  : (G == 1) ? bi  : (G == 2) ? bu  : bo;
    const float* th = (G == 0) ? thf : (G == 1) ? thi : (G == 2) ? thu : tho;
    for (int w = 0; w < NW; ++w) sWh[b][w] = W[r * FIN + EMB + w];
    sB[b]  = bb[r];
    sTh[b] = th[r];
  }
  __syncthreads();

  float h[NW]   = {0.f, 0.f, 0.f, 0.f};
  float cst[NW] = {0.f, 0.f, 0.f, 0.f};
  const float ln16 = logf(16.0f);

  for (int s = 0; s < S_LEN; ++s) {
    const float* p = pre + ((long)s * BATCH + b) * NGATE;
    float pv[NGATE];
    #pragma unroll
    for (int j = 0; j < 4; ++j) {                       // 4x global_load_b128
      const float4 v = reinterpret_cast<const float4*>(p)[j];
      pv[4*j+0] = v.x; pv[4*j+1] = v.y; pv[4*j+2] = v.z; pv[4*j+3] = v.w;
    }

    float q[4][4];
    #pragma unroll
    for (int G = 0; G < 4; ++G) {
      float cz[NW];
      #pragma unroll
      for (int w = 0; w < NW; ++w) {
        const int g = G * 4 + w;
        float z = pv[g] + sB[g];
        #pragma unroll
        for (int u = 0; u < NW; ++u) z = fmaf(h[u], sWh[g][u], z);
        cz[w] = __cosf(z + sTh[g]);                     // v_cos_f32
      }
      q[G][0] = cz[1] * cz[2] * cz[3];
      q[G][1] = cz[0] * cz[1];
      q[G][2] = q[G][1] * cz[2];
      q[G][3] = q[G][2] * cz[3];
    }

    float lg = 0.0f;
    #pragma unroll
    for (int w = 0; w < NW; ++w) {
      const float fg = sig_fast(q[0][w]);
      const float ig = sig_fast(q[1][w]);
      const float gg = tanh_fast(q[2][w]);
      const float og = sig_fast(q[3][w]);
      cst[w] = fmaf(fg, cst[w], ig * gg);
      h[w]   = og * tanh_fast(cst[w]);
      lg += (1.0f + __sinf(h[w])) * 0.5f;
    }
    lg *= 0.25f;

    const float val = lg - (lg + ln16);                 // log_softmax of const row
    const float4 o4 = make_float4(val, val, val, val);
    float4* orow = reinterpret_cast<float4*>(out + ((long)s * BATCH + b) * 16);
    #pragma unroll
    for (int j = 0; j < 4; ++j) orow[j] = o4;           // 4x global_store_b128
  }
}

// ---------------------------------------------------------------------------
extern "C" void kernel_launch(void* const* d_in, const int* in_sizes, int n_in,
                              void* d_out, int out_size, void* d_ws, size_t ws_size,
                              hipStream_t stream) {
  const int*   sentence = (const int*)  d_in[0];
  const float* emb      = (const float*)d_in[1];
  const float* Wf = (const float*)d_in[2];  const float* bf = (const float*)d_in[3];
  const float* Wi = (const float*)d_in[4];  const float* bi = (const float*)d_in[5];
  const float* Wu = (const float*)d_in[6];  const float* bu = (const float*)d_in[7];
  const float* Wo = (const float*)d_in[8];  const float* bo = (const float*)d_in[9];
  const float* thf = (const float*)d_in[10]; const float* thi = (const float*)d_in[11];
  const float* thu = (const float*)d_in[12]; const float* tho = (const float*)d_in[13];
  float* out = (float*)d_out;
  float* pre = (float*)d_ws;     // S*B*16 floats = 2 MB scratch

  dim3 g1(BATCH / 16, S_LEN);
  qlstm_gemm_embed<<<g1, 32, 0, stream>>>(sentence, emb, Wf, Wi, Wu, Wo, pre);
  qlstm_recur<<<1, BATCH, 0, stream>>>(pre, Wf, bf, Wi, bi, Wu, bu, Wo, bo,
                                       thf, thi, thu, tho, out);
}